// PWLU_84756884619350
// MI455X (gfx1250) — compile-verified
//
#include <hip/hip_runtime.h>

// PWLU: channelwise piecewise-linear unit.
//   x      : [32, 256, 56, 56] fp32
//   points : [256, 7] fp32
//   out    : same shape as x
//
// Memory-bound: ~205 MB traffic -> ~8.8 us floor @ 23.3 TB/s. No matrix
// contraction exists, so WMMA is inapplicable; the win is on CDNA5's
// data-movement paths: b128 vector loads, NT b128 stores, LDS-staged
// interleaved {left,diff} table (one ds_load_b64 per element), and
// global_prefetch_b8 for the next grid-stride iteration.

typedef __attribute__((ext_vector_type(4))) float v4f;
typedef __attribute__((ext_vector_type(2))) float v2f;

#define PWLU_BOUND   2.7f
#define PWLU_NREG    6
#define PWLU_NPTS    7
#define PWLU_NCH     256
#define PWLU_HW      3136u      // 56*56, divisible by 4
#define PWLU_THREADS 256        // 8 wave32 waves per block; == N_CHANNELS

__global__ __launch_bounds__(PWLU_THREADS) void pwlu_kernel(
    const float* __restrict__ x,
    const float* __restrict__ points,
    float* __restrict__ out,
    unsigned int nvec,        // number of float4 elements
    unsigned int ntotal)      // total scalar elements (tail safety)
{
    // ---- Interleaved per-channel table in LDS: s_tab[c*6+r] = {left, diff} ----
    // 256 ch * 6 regions * 8 B = 12 KB (vs 320 KB/WGP -> occupancy unconstrained).
    __shared__ v2f s_tab[PWLU_NCH * PWLU_NREG];

    {
        // thread id == channel id: no division in the fill loop.
        int c = (int)threadIdx.x;                   // 0..255
        float p[PWLU_NPTS];
        #pragma unroll
        for (int k = 0; k < PWLU_NPTS; ++k)
            p[k] = points[c * PWLU_NPTS + k];
        #pragma unroll
        for (int r = 0; r < PWLU_NREG; ++r) {
            v2f e;
            e.x = p[r];                             // left point
            e.y = p[r + 1] - p[r];                  // segment slope * region_len
            s_tab[c * PWLU_NREG + r] = e;           // ds_store_b64
        }
    }
    __syncthreads();

    const float inv_len = (float)PWLU_NREG / (2.0f * PWLU_BOUND); // 1/region_length
    const float shift   = PWLU_BOUND * inv_len;                   // BOUND/region_length
    const float rmax    = (float)(PWLU_NREG - 1);

    const v4f* __restrict__ x4 = reinterpret_cast<const v4f*>(x);
    v4f* __restrict__ o4       = reinterpret_cast<v4f*>(out);

    const unsigned int stride = gridDim.x * PWLU_THREADS;

    for (unsigned int j = blockIdx.x * PWLU_THREADS + threadIdx.x; j < nvec; j += stride) {
        // Prefetch this thread's next iteration line (global_prefetch_b8).
        if (j + stride < nvec) {
            __builtin_prefetch((const void*)&x4[j + stride], 0, 0);
        }

        v4f v = x4[j];                              // global_load_b128

        unsigned int i = j * 4u;                    // scalar element index (fits u32)
        unsigned int c = (i / PWLU_HW) & (PWLU_NCH - 1u);
        int base = (int)(c * PWLU_NREG);

        v4f res;
        #pragma unroll
        for (int k = 0; k < 4; ++k) {
            float xv = v[k];
            float xn = fmaf(xv, inv_len, shift);            // (x + BOUND) / region_len
            float cl = fminf(fmaxf(xn, 0.0f), rmax);        // clip to [0, 5]
            int   rg = (int)cl;                             // trunc == floor (cl >= 0)
            float d  = xn - (float)rg;                      // unclamped distance
            v2f   e  = s_tab[base + rg];                    // ds_load_b64 gather
            res[k] = fmaf(d, e.y, e.x);
        }

        __builtin_nontemporal_store(res, &o4[j]);   // global_store_b128 th:NT
    }

    // ---- Scalar tail (ntotal % 4 != 0); not hit for the reference shapes ----
    unsigned int tail = nvec * 4u;
    if (blockIdx.x == 0) {
        for (unsigned int i = tail + threadIdx.x; i < ntotal; i += PWLU_THREADS) {
            unsigned int c = (i / PWLU_HW) & (PWLU_NCH - 1u);
            int base = (int)(c * PWLU_NREG);
            float xv = x[i];
            float xn = fmaf(xv, inv_len, shift);
            float cl = fminf(fmaxf(xn, 0.0f), rmax);
            int   rg = (int)cl;
            float d  = xn - (float)rg;
            v2f   e  = s_tab[base + rg];
            out[i] = fmaf(d, e.y, e.x);
        }
    }
}

extern "C" void kernel_launch(void* const* d_in, const int* in_sizes, int n_in,
                              void* d_out, int out_size, void* d_ws, size_t ws_size,
                              hipStream_t stream) {
    (void)n_in; (void)in_sizes; (void)d_ws; (void)ws_size;

    const float* x      = (const float*)d_in[0];   // [32,256,56,56] fp32
    const float* points = (const float*)d_in[1];   // [256,7] fp32
    float* out          = (float*)d_out;

    unsigned int ntotal = (unsigned int)out_size;  // 25,690,112
    unsigned int nvec   = ntotal / 4u;             // 6,422,528 float4s

    // Grid-stride: ~4096 blocks x 8 waves saturates every WGP while
    // amortizing the per-block LDS table fill (~6 float4s per thread).
    unsigned int blocks = (nvec + PWLU_THREADS - 1) / PWLU_THREADS;
    if (blocks > 4096u) blocks = 4096u;
    if (blocks == 0u)   blocks = 1u;

    pwlu_kernel<<<dim3(blocks), dim3(PWLU_THREADS), 0, stream>>>(
        x, points, out, nvec, ntotal);
}